// EdgeGATBlock_73220602462592
// MI455X (gfx1250) — compile-verified
//
#include <hip/hip_runtime.h>
#include <hip/hip_bf16.h>
#include <stdint.h>

#define HID   256
#define EDIM  64
#define HEADS 8
#define CCH   32
#define NEG_SLOPE 0.2f
#define LN_EPS 1e-5f

typedef float v2f __attribute__((ext_vector_type(2)));
typedef float v8f __attribute__((ext_vector_type(8)));

// ---------- order-preserving float<->uint encoding for atomic segment-max ----
__device__ __forceinline__ unsigned enc_f32(float f) {
    unsigned b = __float_as_uint(f);
    return (b & 0x80000000u) ? ~b : (b | 0x80000000u);
}
__device__ __forceinline__ float dec_f32(unsigned u) {
    unsigned b = (u & 0x80000000u) ? (u & 0x7FFFFFFFu) : ~u;
    return __uint_as_float(b);
}

// ---------------------------------------------------------------------------
// K1: w_eh[d][h] = sum_c W_e[d, h*32+c] * att_edge[h, c]   (64x8 tiny matrix)
// ---------------------------------------------------------------------------
__global__ void k_weh(const float* __restrict__ W_e,
                      const float* __restrict__ att_edge,
                      float* __restrict__ w_eh) {
    int d = blockIdx.x * blockDim.x + threadIdx.x;
    if (d >= EDIM) return;
    #pragma unroll
    for (int h = 0; h < HEADS; ++h) {
        float s = 0.f;
        #pragma unroll
        for (int c = 0; c < CCH; ++c)
            s += W_e[d * HID + h * CCH + c] * att_edge[h * CCH + c];
        w_eh[d * HEADS + h] = s;
    }
}

// ---------------------------------------------------------------------------
// K2: xp = x @ W via V_WMMA_F32_16X16X4_F32.
// Block = 256 threads = 8 waves. Wave handles a 16x64 output tile:
//   wave>>2 selects 16-row subtile, wave&3 selects 64-col band.
// Block covers 32 rows x 256 cols. K-loop: 256 in steps of 4.
// A frag (16x4 f32): lanes 0-15 M=0..15 {K=k,k+1}; lanes 16-31 {K=k+2,k+3}.
// B frag (4x16 f32): lanes 0-15 rows K=k,k+1; lanes 16-31 rows K=k+2,k+3.
// C (16x16 f32): VGPR i: lanes0-15 -> M=i, lanes16-31 -> M=8+i; N=lane&15.
// ---------------------------------------------------------------------------
__global__ void __launch_bounds__(256)
k_node_proj(const float* __restrict__ x, const float* __restrict__ W,
            float* __restrict__ xp, int n_nodes) {
    const int wave = threadIdx.x >> 5;
    const int lane = threadIdx.x & 31;
    const int row_base = (blockIdx.x * 2 + (wave >> 2)) * 16;
    const int col_base = (wave & 3) * 64;
    if (row_base >= n_nodes) return;          // wave-uniform: EXEC stays all-1s

    int m = lane & 15;
    int kk = (lane >> 4) * 2;                  // 0 or 2
    int row = row_base + m;
    if (row >= n_nodes) row = n_nodes - 1;     // clamp (dup) to keep EXEC full
    const float* xrow = x + (size_t)row * HID;
    const int ncol = col_base + (lane & 15);

    v8f acc0 = {}, acc1 = {}, acc2 = {}, acc3 = {};
    for (int k = 0; k < HID; k += 4) {
        v2f a;
        a.x = xrow[k + kk];
        a.y = xrow[k + kk + 1];
        const float* w0 = W + (size_t)(k + kk) * HID;
        const float* w1 = W + (size_t)(k + kk + 1) * HID;
        v2f b0, b1, b2, b3;
        b0.x = w0[ncol];      b0.y = w1[ncol];
        b1.x = w0[ncol + 16]; b1.y = w1[ncol + 16];
        b2.x = w0[ncol + 32]; b2.y = w1[ncol + 32];
        b3.x = w0[ncol + 48]; b3.y = w1[ncol + 48];
        acc0 = __builtin_amdgcn_wmma_f32_16x16x4_f32(false, a, false, b0, (short)0, acc0, false, false);
        acc1 = __builtin_amdgcn_wmma_f32_16x16x4_f32(false, a, false, b1, (short)0, acc1, false, false);
        acc2 = __builtin_amdgcn_wmma_f32_16x16x4_f32(false, a, false, b2, (short)0, acc2, false, false);
        acc3 = __builtin_amdgcn_wmma_f32_16x16x4_f32(false, a, false, b3, (short)0, acc3, false, false);
    }

    const int srow = row_base + ((lane >> 4) * 8);
    const int scol = col_base + (lane & 15);
    #pragma unroll
    for (int i = 0; i < 8; ++i) {
        int r = srow + i;
        if (r < n_nodes) {
            float* o = xp + (size_t)r * HID + scol;
            o[0]  = acc0[i];
            o[16] = acc1[i];
            o[32] = acc2[i];
            o[48] = acc3[i];
        }
    }
}

// ---------------------------------------------------------------------------
// K3: a_src[n,h], a_dst[n,h] from xp (one wave per node, lane covers 8 cols)
// ---------------------------------------------------------------------------
__global__ void __launch_bounds__(256)
k_att_dots(const float* __restrict__ xp, const float* __restrict__ att_src,
           const float* __restrict__ att_dst, float* __restrict__ a_src,
           float* __restrict__ a_dst, int n_nodes) {
    const int wave = threadIdx.x >> 5;
    const int lane = threadIdx.x & 31;
    const int node = blockIdx.x * 8 + wave;
    if (node >= n_nodes) return;
    const float* row = xp + (size_t)node * HID + lane * 8;
    float s = 0.f, t = 0.f;
    #pragma unroll
    for (int j = 0; j < 8; ++j) {
        float v = row[j];
        int col = lane * 8 + j;
        s += v * att_src[col];
        t += v * att_dst[col];
    }
    // lanes 4k..4k+3 cover cols of head k -> reduce within groups of 4
    s += __shfl_xor(s, 1); s += __shfl_xor(s, 2);
    t += __shfl_xor(t, 1); t += __shfl_xor(t, 2);
    if ((lane & 3) == 0) {
        a_src[node * HEADS + (lane >> 2)] = s;
        a_dst[node * HEADS + (lane >> 2)] = t;
    }
}

// ---------------------------------------------------------------------------
// K4: per-edge logit + LeakyReLU + atomic segment max (encoded uint).
// One wave per edge; lane = h*4+q, 4 lanes reduce 64 edge features per head.
// ---------------------------------------------------------------------------
__global__ void __launch_bounds__(256)
k_alpha(const float* __restrict__ edge_attr, const int* __restrict__ ei,
        const float* __restrict__ w_eh, const float* __restrict__ a_src,
        const float* __restrict__ a_dst, float* __restrict__ alpha,
        unsigned* __restrict__ menc, int n_edges) {
    const int wave = threadIdx.x >> 5;
    const int lane = threadIdx.x & 31;
    const int e = blockIdx.x * 8 + wave;
    if (e >= n_edges) return;
    const int h = lane >> 2, q = lane & 3;
    const int src = ei[e];
    const int dst = ei[n_edges + e];
    const float* ea = edge_attr + (size_t)e * EDIM + q * 16;
    float s = 0.f;
    #pragma unroll
    for (int j = 0; j < 16; ++j)
        s += ea[j] * w_eh[(q * 16 + j) * HEADS + h];
    s += __shfl_xor(s, 1);
    s += __shfl_xor(s, 2);
    if (q == 0) {
        float al = a_src[src * HEADS + h] + a_dst[dst * HEADS + h] + s;
        al = (al > 0.f) ? al : NEG_SLOPE * al;
        alpha[(size_t)e * HEADS + h] = al;
        atomicMax(&menc[dst * HEADS + h], enc_f32(al));
    }
}

// ---------------------------------------------------------------------------
// K5: ea = exp(alpha - m[dst]); atomicAdd den; atomicAdd UNNORMALIZED message
// acc[dst] += ea * xp[src].  (1/den folded into finalize.)  Wave per edge.
// ---------------------------------------------------------------------------
__global__ void __launch_bounds__(256)
k_scatter(const float* __restrict__ xp, const int* __restrict__ ei,
          const float* __restrict__ alpha, const unsigned* __restrict__ menc,
          float* __restrict__ den, float* __restrict__ acc, int n_edges) {
    const int wave = threadIdx.x >> 5;
    const int lane = threadIdx.x & 31;
    const int e = blockIdx.x * 8 + wave;
    if (e >= n_edges) return;
    const int src = ei[e];
    const int dst = ei[n_edges + e];

    if (lane < HEADS) {
        float m = dec_f32(menc[dst * HEADS + lane]);
        float ea = __expf(alpha[(size_t)e * HEADS + lane] - m);
        atomicAdd(&den[dst * HEADS + lane], ea);
    }
    const float* xs = xp + (size_t)src * HID;
    float* ad = acc + (size_t)dst * HID;
    #pragma unroll
    for (int h = 0; h < HEADS; ++h) {
        float m = dec_f32(menc[dst * HEADS + h]);
        float ea = __expf(alpha[(size_t)e * HEADS + h] - m);
        atomicAdd(&ad[h * CCH + lane], ea * xs[h * CCH + lane]);
    }
}

// ---------------------------------------------------------------------------
// K6: out = LN(x + acc/den + bias) * gamma + beta.  Block (256 thr) per node.
// ---------------------------------------------------------------------------
__global__ void __launch_bounds__(256)
k_finalize(const float* __restrict__ x, const float* __restrict__ acc,
           const float* __restrict__ den, const float* __restrict__ bias,
           const float* __restrict__ gamma, const float* __restrict__ beta,
           float* __restrict__ out, int n_nodes) {
    const int n = blockIdx.x;
    if (n >= n_nodes) return;
    const int t = threadIdx.x;
    const int h = t >> 5;
    float v = acc[(size_t)n * HID + t] / (den[n * HEADS + h] + 1e-16f);
    float y = x[(size_t)n * HID + t] + v + bias[t];

    float s = y, s2 = y * y;
    #pragma unroll
    for (int off = 16; off >= 1; off >>= 1) {
        s  += __shfl_xor(s, off);
        s2 += __shfl_xor(s2, off);
    }
    __shared__ float ws1[8], ws2[8];
    const int lane = t & 31, wave = t >> 5;
    if (lane == 0) { ws1[wave] = s; ws2[wave] = s2; }
    __syncthreads();
    if (wave == 0) {
        float a = (lane < 8) ? ws1[lane] : 0.f;
        float b = (lane < 8) ? ws2[lane] : 0.f;
        #pragma unroll
        for (int off = 4; off >= 1; off >>= 1) {
            a += __shfl_xor(a, off);
            b += __shfl_xor(b, off);
        }
        if (lane == 0) { ws1[0] = a; ws2[0] = b; }
    }
    __syncthreads();
    float mean = ws1[0] * (1.f / HID);
    float var  = ws2[0] * (1.f / HID) - mean * mean;
    out[(size_t)n * HID + t] = (y - mean) * rsqrtf(var + LN_EPS) * gamma[t] + beta[t];
}

// ---------------------------------------------------------------------------
extern "C" void kernel_launch(void* const* d_in, const int* in_sizes, int n_in,
                              void* d_out, int out_size, void* d_ws, size_t ws_size,
                              hipStream_t stream) {
    const float* x         = (const float*)d_in[0];
    const int*   ei        = (const int*)  d_in[1];
    const float* edge_attr = (const float*)d_in[2];
    const float* W         = (const float*)d_in[3];
    const float* W_e       = (const float*)d_in[4];
    const float* att_src   = (const float*)d_in[5];
    const float* att_dst   = (const float*)d_in[6];
    const float* att_edge  = (const float*)d_in[7];
    const float* bias      = (const float*)d_in[8];
    const float* gamma     = (const float*)d_in[9];
    const float* beta      = (const float*)d_in[10];
    float* out = (float*)d_out;

    const int n_nodes = in_sizes[0] / HID;
    const int n_edges = in_sizes[1] / 2;

    // ---- carve workspace (256B aligned) ----
    char* ws = (char*)d_ws;
    size_t off = 0;
    auto carve = [&](size_t bytes) -> char* {
        char* p = ws + off;
        off += (bytes + 255) & ~(size_t)255;
        return p;
    };
    float*    xp    = (float*)   carve((size_t)n_nodes * HID   * 4);
    float*    acc   = (float*)   carve((size_t)n_nodes * HID   * 4);
    float*    alpha = (float*)   carve((size_t)n_edges * HEADS * 4);
    float*    a_src = (float*)   carve((size_t)n_nodes * HEADS * 4);
    float*    a_dst = (float*)   carve((size_t)n_nodes * HEADS * 4);
    float*    den   = (float*)   carve((size_t)n_nodes * HEADS * 4);
    unsigned* menc  = (unsigned*)carve((size_t)n_nodes * HEADS * 4);
    float*    w_eh  = (float*)   carve((size_t)EDIM * HEADS * 4);
    (void)ws_size;

    // zero-init: acc, den, and encoded max (encoded 0 < enc(any finite float))
    hipMemsetAsync(acc,  0, (size_t)n_nodes * HID   * 4, stream);
    hipMemsetAsync(den,  0, (size_t)n_nodes * HEADS * 4, stream);
    hipMemsetAsync(menc, 0, (size_t)n_nodes * HEADS * 4, stream);

    k_weh<<<1, 64, 0, stream>>>(W_e, att_edge, w_eh);

    k_node_proj<<<(n_nodes + 31) / 32, 256, 0, stream>>>(x, W, xp, n_nodes);

    k_att_dots<<<(n_nodes + 7) / 8, 256, 0, stream>>>(xp, att_src, att_dst,
                                                      a_src, a_dst, n_nodes);

    k_alpha<<<(n_edges + 7) / 8, 256, 0, stream>>>(edge_attr, ei, w_eh,
                                                   a_src, a_dst, alpha, menc, n_edges);

    k_scatter<<<(n_edges + 7) / 8, 256, 0, stream>>>(xp, ei, alpha, menc,
                                                     den, acc, n_edges);

    k_finalize<<<n_nodes, 256, 0, stream>>>(x, acc, den, bias, gamma, beta,
                                            out, n_nodes);
}